// WassersteinHistogramLoss_37254546325519
// MI455X (gfx1250) — compile-verified
//
#include <hip/hip_runtime.h>

#define NUM_BINS 256
#define NBATCH 32
#define PIX_PER_IMG (3 * 512 * 512)            // 786432 per image
#define V4_PER_IMG (PIX_PER_IMG / 4)           // 196608 float4 per image
#define BLOCKS_PER_IMG 96
#define V4_PER_BLOCK (V4_PER_IMG / BLOCKS_PER_IMG)  // 2048 float4 per block
#define HIST_THREADS 256

typedef __attribute__((ext_vector_type(2))) float v2f;
typedef __attribute__((ext_vector_type(8))) float v8f;

// ---------------------------------------------------------------------------
// Kernel 0: zero the global histogram workspace (harness does not re-poison).
// ---------------------------------------------------------------------------
__global__ void zero_hist_kernel(unsigned* __restrict__ h) {
    int i = blockIdx.x * blockDim.x + threadIdx.x;
    if (i < NBATCH * NUM_BINS) h[i] = 0u;
}

// Replicate JAX bin math exactly: d = x*0.5+0.5 (mul,add — no FMA), *255, trunc.
__device__ __forceinline__ int pix_bin(float x) {
    float d = __fadd_rn(__fmul_rn(x, 0.5f), 0.5f);
    int idx = (int)__fmul_rn(d, 255.0f);
    return idx < 0 ? 0 : (idx > 255 ? 255 : idx);
}

// ---------------------------------------------------------------------------
// Kernel 1: bandwidth-bound histogram. float4 loads (global_load_b128),
// per-wave private LDS histograms (wave32 -> 8 waves per 256-thread block),
// integer atomics only (bit-exact => deterministic).
// ---------------------------------------------------------------------------
__global__ void __launch_bounds__(HIST_THREADS)
hist_kernel(const float* __restrict__ imgs, unsigned* __restrict__ ghist) {
    __shared__ unsigned sh[8][NUM_BINS];   // 8 KB: one 256-bin hist per wave
    const int tid  = threadIdx.x;
    const int wave = tid >> 5;             // wave32
    const int img  = blockIdx.x / BLOCKS_PER_IMG;
    const int blk  = blockIdx.x % BLOCKS_PER_IMG;

    for (int i = tid; i < 8 * NUM_BINS; i += HIST_THREADS) (&sh[0][0])[i] = 0u;
    __syncthreads();

    const float4* p = (const float4*)imgs + (size_t)img * V4_PER_IMG
                                          + (size_t)blk * V4_PER_BLOCK;
    for (int i = tid; i < V4_PER_BLOCK; i += HIST_THREADS) {
        float4 v = p[i];
        atomicAdd(&sh[wave][pix_bin(v.x)], 1u);
        atomicAdd(&sh[wave][pix_bin(v.y)], 1u);
        atomicAdd(&sh[wave][pix_bin(v.z)], 1u);
        atomicAdd(&sh[wave][pix_bin(v.w)], 1u);
    }
    __syncthreads();

    if (tid < NUM_BINS) {
        unsigned s = 0;
#pragma unroll
        for (int w = 0; w < 8; ++w) s += sh[w][tid];
        if (s) atomicAdd(&ghist[img * NUM_BINS + tid], s);
    }
}

// ---------------------------------------------------------------------------
// Kernel 2: finalize. One block, 32 waves (1024 threads).
//   diff_p = gen_hist/sum - tgt_hist/sum  (per batch row, in LDS)
//   cdf diff = diff_p (32x256) x upper-triangular-ones (256x256)
//     -> tiled on V_WMMA_F32_16X16X4_F32, one 16x16 output tile per wave,
//        triangular structure skips all-zero K tiles. Tile coords pinned to
//        SGPRs via readfirstlane so the K loop is uniform (EXEC stays all-1s
//        across every WMMA, per ISA requirement).
//   loss = mean |.| via fixed-order reductions (deterministic).
// ---------------------------------------------------------------------------
__global__ void __launch_bounds__(1024)
finalize_kernel(const unsigned* __restrict__ ghist,
                const float* __restrict__ tgt,
                float* __restrict__ out) {
    __shared__ float dp[NBATCH * NUM_BINS];   // 32 KB
    __shared__ float partials[32];

    const int tid  = threadIdx.x;
    const int wid  = tid >> 5;    // wave id 0..31
    const int lane = tid & 31;

    // --- per-batch normalization: wave w owns batch row w ---
    {
        const unsigned* gh = ghist + wid * NUM_BINS;
        const float*    th = tgt   + wid * NUM_BINS;
        float gs = 0.0f, ts = 0.0f;
        for (int j = lane; j < NUM_BINS; j += 32) {
            gs += (float)gh[j];
            ts += th[j];
        }
#pragma unroll
        for (int off = 16; off >= 1; off >>= 1) {
            gs += __shfl_down(gs, off, 32);
            ts += __shfl_down(ts, off, 32);
        }
        gs = __shfl(gs, 0, 32);
        ts = __shfl(ts, 0, 32);
        const float gin = 1.0f / (gs + 1e-8f);
        const float tin = 1.0f / (ts + 1e-8f);
        for (int j = lane; j < NUM_BINS; j += 32)
            dp[wid * NUM_BINS + j] = (float)gh[j] * gin - th[j] * tin;
    }
    __syncthreads();

    // --- WMMA cumsum: output tile (m,n); m = wid/16, n = wid%16 ---
    // wid is wave-uniform; force it into SGPRs so the loop below is uniform.
    const int m  = __builtin_amdgcn_readfirstlane(wid >> 4);  // batch-row tile
    const int n  = __builtin_amdgcn_readfirstlane(wid & 15);  // bin-col tile
    const int ln = lane & 15;         // position within 16 (M for A, N for B)
    const int kh = lane >> 4;         // K half-select (K = 2*kh + {0,1})

    v8f c = {};
    const int col   = n * 16 + ln;    // global output column for B predicate
    const int ktmax = 4 * n + 3;      // triangular: higher K tiles are all-zero
    for (int kt = 0; kt <= ktmax; ++kt) {
        const int k0 = kt * 4 + kh * 2;
        v2f a, b;
        a.x = dp[(m * 16 + ln) * NUM_BINS + k0];
        a.y = dp[(m * 16 + ln) * NUM_BINS + k0 + 1];
        b.x = (k0     <= col) ? 1.0f : 0.0f;   // T[k][col] = (k <= col)
        b.y = (k0 + 1 <= col) ? 1.0f : 0.0f;
        c = __builtin_amdgcn_wmma_f32_16x16x4_f32(
                /*neg_a=*/false, a, /*neg_b=*/false, b,
                /*c_mod=*/(short)0, c, /*reuse_a=*/false, /*reuse_b=*/false);
    }

    // --- |.| and fixed-order reductions ---
    float s = 0.0f;
#pragma unroll
    for (int r = 0; r < 8; ++r) s += fabsf(c[r]);
#pragma unroll
    for (int off = 16; off >= 1; off >>= 1) s += __shfl_down(s, off, 32);
    if (lane == 0) partials[wid] = s;
    __syncthreads();

    if (tid == 0) {
        float t = 0.0f;
        for (int i = 0; i < 32; ++i) t += partials[i];
        out[0] = t / (float)(NBATCH * NUM_BINS);
    }
}

// ---------------------------------------------------------------------------
extern "C" void kernel_launch(void* const* d_in, const int* in_sizes, int n_in,
                              void* d_out, int out_size, void* d_ws, size_t ws_size,
                              hipStream_t stream) {
    const float* imgs = (const float*)d_in[0];   // [32,3,512,512] fp32
    const float* tgt  = (const float*)d_in[1];   // [32,256] fp32
    float* out        = (float*)d_out;           // scalar fp32
    unsigned* ghist   = (unsigned*)d_ws;         // 32*256 u32 = 32 KB

    (void)in_sizes; (void)n_in; (void)out_size; (void)ws_size;

    zero_hist_kernel<<<(NBATCH * NUM_BINS + 255) / 256, 256, 0, stream>>>(ghist);
    hist_kernel<<<NBATCH * BLOCKS_PER_IMG, HIST_THREADS, 0, stream>>>(imgs, ghist);
    finalize_kernel<<<1, 1024, 0, stream>>>(ghist, tgt, out);
}